// S4_PTD_Kernel_76630806495599
// MI455X (gfx1250) — compile-verified
//
#include <hip/hip_runtime.h>
#include <hip/hip_bf16.h>

// S4D Vandermonde kernel: K[h,l] = 2*Re( sum_n c'[h,n] * exp(dtA[h,n])^l )
// H=1024 heads, NH=32 modes, L=4096 steps.
//
// Reformulated per head as a real GEMM  K[h] (256x16) = A'(256x64) @ B'(64x16)
// using the power factorization l = 256*j2 + 16*j1 + l0, executed with
// V_WMMA_F32_16X16X4_F32 (full fp32 precision; kernel is store-BW bound so
// f32 WMMA rate is not the bottleneck).

#define S4_H   1024
#define S4_NH  32
#define S4_L   4096

typedef __attribute__((ext_vector_type(2))) float v2f;
typedef __attribute__((ext_vector_type(8))) float v8f;

__device__ __forceinline__ float2 cmul(float2 a, float2 b) {
    return make_float2(a.x * b.x - a.y * b.y, a.x * b.y + a.y * b.x);
}

// exp(z * t) for complex z, real t: exp(re*t) * (cos(im*t) + i sin(im*t))
__device__ __forceinline__ float2 cexp_scaled(float2 z, float t) {
    float s, c;
    sincosf(z.y * t, &s, &c);
    float e = expf(z.x * t);
    return make_float2(e * c, e * s);
}

__global__ __launch_bounds__(512)
void s4d_vandermonde_wmma(const float* __restrict__ Cin,
                          const float* __restrict__ log_dt,
                          const float* __restrict__ log_A_real,
                          const float* __restrict__ A_imag,
                          float* __restrict__ out) {
    // Per-head tables (complex, interleaved re/im):
    //   sW[n][j] = w_n^j          (w = exp(dtA)),   j in [0,16)
    //   sV[n][j] = w_n^(16 j)
    //   sU[n][j] = w_n^(256 j)
    //   sCp[n]   = 2 * C_n * dt * (w_n - 1) / dtA_n
    __shared__ float2 sW[S4_NH][16];
    __shared__ float2 sV[S4_NH][16];
    __shared__ float2 sU[S4_NH][16];
    __shared__ float2 sCp[S4_NH];

    const int h   = blockIdx.x;
    const int tid = threadIdx.x;

    // ---------- Phase 1: build tables (512 threads -> (n, j) pairs) ----------
    {
        const int n = tid >> 4;    // mode 0..31
        const int j = tid & 15;    // power index 0..15

        const float dt = expf(log_dt[h]);
        const float ar = -expf(log_A_real[h * S4_NH + n]) * dt;  // Re(dtA) < 0
        const float ai = -A_imag[h * S4_NH + n] * dt;            // Im(dtA)
        const float2 dtA = make_float2(ar, ai);

        sW[n][j] = cexp_scaled(dtA, (float)j);
        sV[n][j] = cexp_scaled(dtA, (float)(16 * j));
        sU[n][j] = cexp_scaled(dtA, (float)(256 * j));

        if (j == 0) {
            // c' = 2 * C * (exp(dtA) - 1) / A,  A = dtA / dt
            //    = 2 * dt * C * (E - 1) * conj(dtA) / |dtA|^2
            float2 E = cexp_scaled(dtA, 1.0f);
            float2 Em1 = make_float2(E.x - 1.0f, E.y);
            float inv2 = 1.0f / (dtA.x * dtA.x + dtA.y * dtA.y);
            float2 invA = make_float2(dtA.x * inv2, -dtA.y * inv2);
            float2 Cc = make_float2(Cin[(h * S4_NH + n) * 2 + 0],
                                    Cin[(h * S4_NH + n) * 2 + 1]);
            float2 cp = cmul(cmul(Cc, Em1), invA);
            cp.x *= 2.0f * dt;
            cp.y *= 2.0f * dt;
            sCp[n] = cp;
        }
    }
    __syncthreads();

    // ---------- Phase 2: per-wave 16x16 output tile via 16 WMMA k-steps ------
    const int wv   = tid >> 5;        // wave id 0..15  -> M-tile (l_hi block)
    const int lane = tid & 31;
    const int hi   = lane >> 4;       // lane half: selects K parity (n = 2s+hi)
    const int lj   = lane & 15;       // row-in-tile (A) / column l0 (B/D)

    v8f acc = {};

    #pragma unroll
    for (int s = 0; s < 16; ++s) {
        const int n = 2 * s + hi;

        // A'[M, 4s+2hi .. 4s+2hi+1] for M = 16*wv + lj :
        //   z = cp[n] * U[n][M>>4] * V[n][M&15];  frag = {Re z, -Im z}
        float2 cpv = sCp[n];
        float2 uu  = sU[n][wv];
        float2 vv  = sV[n][lj];
        float2 z   = cmul(cmul(cpv, uu), vv);

        // B'[4s+2hi .. 4s+2hi+1, l0] = {Re, Im} of w[n]^{l0}
        float2 ww  = sW[n][lj];

        v2f a; a[0] = z.x;  a[1] = -z.y;
        v2f b; b[0] = ww.x; b[1] = ww.y;

        // D = A(16x4,f32) * B(4x16,f32) + C   -> v_wmma_f32_16x16x4_f32
        acc = __builtin_amdgcn_wmma_f32_16x16x4_f32(
            /*neg_a=*/false, a, /*neg_b=*/false, b,
            /*c_mod=*/(short)0, acc, /*reuse_a=*/false, /*reuse_b=*/false);
    }

    // ---------- Store: D VGPR v holds row M = v + 8*hi of the tile ----------
    float* __restrict__ outp = out + (size_t)h * S4_L;
    #pragma unroll
    for (int v = 0; v < 8; ++v) {
        const int l_hi = 16 * wv + v + 8 * hi;     // 0..255
        outp[l_hi * 16 + lj] = acc[v];             // l = 16*l_hi + l0
    }
}

extern "C" void kernel_launch(void* const* d_in, const int* in_sizes, int n_in,
                              void* d_out, int out_size, void* d_ws, size_t ws_size,
                              hipStream_t stream) {
    const float* Cin         = (const float*)d_in[0];  // [H, NH, 2]
    const float* log_dt      = (const float*)d_in[1];  // [H]
    const float* log_A_real  = (const float*)d_in[2];  // [H, NH]
    const float* A_imag      = (const float*)d_in[3];  // [H, NH]
    // d_in[4] is L (scalar int) == 4096, compiled in as S4_L.
    float* out = (float*)d_out;                        // [H, L] f32

    (void)in_sizes; (void)n_in; (void)out_size; (void)d_ws; (void)ws_size;

    dim3 grid(S4_H);
    dim3 block(512);   // 16 wave32s: one 16x16 output tile per wave
    s4d_vandermonde_wmma<<<grid, block, 0, stream>>>(
        Cin, log_dt, log_A_real, A_imag, out);
}